// SimpleSNN_17188459118719
// MI455X (gfx1250) — compile-verified
//
#include <hip/hip_runtime.h>

// CDNA5 wave32 WMMA f32 16x16x4: A (16x4) = 2 VGPRs, B (4x16) = 2 VGPRs, C/D (16x16) = 8 VGPRs.
typedef __attribute__((ext_vector_type(2))) float v2f;
typedef __attribute__((ext_vector_type(8))) float v8f;

#define BETA 0.90483741803595952f   // exp(-1/10)
#define TH   15.0f

// D = A x B + C with:
//   A layout: (m,k) -> VGPR (k%2), lane m + 16*(k/2)
//   B layout: (k,n) -> VGPR (k%2), lane n + 16*(k/2)   [analog of the 16-bit B striping]
//   C/D     : (m,n) -> VGPR (m%8), lane n + 16*(m/8)
__global__ __launch_bounds__(256) void snn_lif3_kernel(
    const float* __restrict__ x,
    const float* __restrict__ mem1, const float* __restrict__ mem2,
    const float* __restrict__ mem3,
    const float* __restrict__ W1, const float* __restrict__ W2,
    const float* __restrict__ W3,
    float* __restrict__ out, int Btotal)
{
    const int lane = threadIdx.x & 31;
    const int wave = threadIdx.x >> 5;
    const long long tile = (long long)blockIdx.x * (blockDim.x >> 5) + wave;
    const long long base = tile * 16;
    const long long n    = base + (lane & 15);
    const bool valid = (lane < 16) && (n < (long long)Btotal);

    // ---- A operands: weights zero-padded to 16x4 (rows>=out_feat and k>=in_feat are 0)
    float a1_0 = 0.f, a1_1 = 0.f;   // W1: [2 out][2 in]
    float a2_0 = 0.f, a2_1 = 0.f;   // W2: [3 out][2 in]
    float a3_0 = 0.f, a3_1 = 0.f;   // W3: [2 out][3 in]
    if (lane < 2) { a1_0 = W1[lane*2+0]; a1_1 = W1[lane*2+1]; }
    if (lane < 3) { a2_0 = W2[lane*2+0]; a2_1 = W2[lane*2+1]; }
    if (lane < 2) { a3_0 = W3[lane*3+0]; a3_1 = W3[lane*3+1]; }
    if (lane >= 16 && lane < 18) { a3_0 = W3[(lane-16)*3+2]; }  // k=2 plane of W3

    // ---- streaming loads (lanes 0-15 carry samples; high lanes must be 0 = K padding)
    float b1_0 = 0.f, b1_1 = 0.f;
    float m1_0 = 0.f, m1_1 = 0.f;
    float m2_0 = 0.f, m2_1 = 0.f, m2_2 = 0.f;
    float m3_0 = 0.f, m3_1 = 0.f;
    if (valid) {
        v2f xv  = *(const v2f*)(x    + n*2);  b1_0 = xv.x;  b1_1 = xv.y;
        v2f m1v = *(const v2f*)(mem1 + n*2);  m1_0 = m1v.x; m1_1 = m1v.y;
        m2_0 = mem2[n*3+0]; m2_1 = mem2[n*3+1]; m2_2 = mem2[n*3+2];
        v2f m3v = *(const v2f*)(mem3 + n*2);  m3_0 = m3v.x; m3_1 = m3v.y;
    }

    // ================= layer 1: cur1 = W1 @ x^T, C folds beta*mem1 =================
    v2f A1 = {a1_0, a1_1};
    v2f B1 = {b1_0, b1_1};
    v8f C1 = {};
    C1[0] = BETA * m1_0;
    C1[1] = BETA * m1_1;
    v8f D1 = __builtin_amdgcn_wmma_f32_16x16x4_f32(false, A1, false, B1,
                                                   (short)0, C1, false, false);
    float nm1_0 = (m1_0 > TH) ? 0.f : D1[0];
    float nm1_1 = (m1_1 > TH) ? 0.f : D1[1];
    float s1_0  = (nm1_0 > TH) ? 1.f : 0.f;
    float s1_1  = (nm1_1 > TH) ? 1.f : 0.f;

    // ================= layer 2: spk1 is already in B layout (features 0,1) =========
    v2f A2 = {a2_0, a2_1};
    v2f B2 = {s1_0, s1_1};            // high lanes are 0 by construction (K=2,3 pad)
    v8f C2 = {};
    C2[0] = BETA * m2_0;
    C2[1] = BETA * m2_1;
    C2[2] = BETA * m2_2;
    v8f D2 = __builtin_amdgcn_wmma_f32_16x16x4_f32(false, A2, false, B2,
                                                   (short)0, C2, false, false);
    float nm2_0 = (m2_0 > TH) ? 0.f : D2[0];
    float nm2_1 = (m2_1 > TH) ? 0.f : D2[1];
    float nm2_2 = (m2_2 > TH) ? 0.f : D2[2];
    float s2_0  = (nm2_0 > TH) ? 1.f : 0.f;
    float s2_1  = (nm2_1 > TH) ? 1.f : 0.f;
    float s2_2  = (nm2_2 > TH) ? 1.f : 0.f;

    // ================= layer 3: K=3 input, feature 2 moves to high half-lanes ======
    float s2_2_bcast = __shfl(s2_2, lane & 15, 32);     // lane l reads lane l-16's value
    float b3_0 = (lane < 16) ? s2_0 : s2_2_bcast;       // K=0 | K=2
    float b3_1 = (lane < 16) ? s2_1 : 0.f;              // K=1 | K=3 (pad)
    v2f A3 = {a3_0, a3_1};
    v2f B3 = {b3_0, b3_1};
    v8f C3 = {};
    C3[0] = BETA * m3_0;
    C3[1] = BETA * m3_1;
    v8f D3 = __builtin_amdgcn_wmma_f32_16x16x4_f32(false, A3, false, B3,
                                                   (short)0, C3, false, false);
    float nm3_0 = (m3_0 > TH) ? 0.f : D3[0];
    float nm3_1 = (m3_1 > TH) ? 0.f : D3[1];
    float s3_0  = (nm3_0 > TH) ? 1.f : 0.f;
    float s3_1  = (nm3_1 > TH) ? 1.f : 0.f;

    // ---- stores: out = [spk3 (2B) | mem1 (2B) | mem2 (3B) | mem3 (2B)]
    if (valid) {
        const long long Bt = (long long)Btotal;
        *(v2f*)(out + n*2)            = (v2f){s3_0,  s3_1};
        *(v2f*)(out + Bt*2 + n*2)     = (v2f){nm1_0, nm1_1};
        float* o2 = out + Bt*4 + n*3;
        o2[0] = nm2_0; o2[1] = nm2_1; o2[2] = nm2_2;
        *(v2f*)(out + Bt*7 + n*2)     = (v2f){nm3_0, nm3_1};
    }
}

extern "C" void kernel_launch(void* const* d_in, const int* in_sizes, int n_in,
                              void* d_out, int out_size, void* d_ws, size_t ws_size,
                              hipStream_t stream) {
    const float* x    = (const float*)d_in[0];
    const float* mem1 = (const float*)d_in[1];
    const float* mem2 = (const float*)d_in[2];
    const float* mem3 = (const float*)d_in[3];
    const float* W1   = (const float*)d_in[4];
    const float* W2   = (const float*)d_in[5];
    const float* W3   = (const float*)d_in[6];
    float* out = (float*)d_out;

    const int Btotal = in_sizes[0] / 2;          // x is [B, 2]
    const int samplesPerBlock = 128;             // 8 waves * 16 samples
    const int grid = (Btotal + samplesPerBlock - 1) / samplesPerBlock;
    snn_lif3_kernel<<<grid, 256, 0, stream>>>(x, mem1, mem2, mem3, W1, W2, W3,
                                              out, Btotal);
}